// ReparameterizedGMM_Linear_Multiclass_27539330302048
// MI455X (gfx1250) — compile-verified
//
#include <hip/hip_runtime.h>

// Problem constants (from the reference)
#define N_Z       128
#define N_COMP    10
#define N_CLASSES 10
#define BS_TOT    8192
#define N_BINS    (N_CLASSES * N_COMP)   // 100 distinct (class, component) matrices

typedef __attribute__((ext_vector_type(2))) float v2f;
typedef __attribute__((ext_vector_type(8))) float v8f;

// ---------------------------------------------------------------------------
// Kernel 1: per-sample hard gumbel argmax -> bin id, histogram counts.
// ---------------------------------------------------------------------------
__global__ void bin_count_kernel(const float* __restrict__ gumbel,   // (C, BS, K)
                                 const float* __restrict__ logits,   // (C, K)
                                 const int*   __restrict__ y,        // (BS,)
                                 int* __restrict__ counts,
                                 int* __restrict__ binOf) {
  int b = blockIdx.x * blockDim.x + threadIdx.x;
  if (b >= BS_TOT) return;
  int c = y[b];
  const float* g  = gumbel + ((size_t)c * BS_TOT + b) * N_COMP;
  const float* lg = logits + c * N_COMP;
  float best = lg[0] + g[0];
  int bk = 0;
#pragma unroll
  for (int k = 1; k < N_COMP; ++k) {
    float s = lg[k] + g[k];
    if (s > best) { best = s; bk = k; }   // strict '>' keeps first max, like jnp.argmax
  }
  int bin = c * N_COMP + bk;
  binOf[b] = bin;
  atomicAdd(&counts[bin], 1);
}

// ---------------------------------------------------------------------------
// Kernel 2: tiny exclusive scan over 100 bins (serial; trivial size).
// ---------------------------------------------------------------------------
__global__ void scan_kernel(const int* __restrict__ counts,
                            int* __restrict__ offsets,
                            int* __restrict__ cursors) {
  if (threadIdx.x == 0 && blockIdx.x == 0) {
    int acc = 0;
    for (int i = 0; i < N_BINS; ++i) {
      offsets[i] = acc;
      cursors[i] = acc;
      acc += counts[i];
    }
  }
}

// ---------------------------------------------------------------------------
// Kernel 3: scatter sample indices into compact per-bin lists.
// ---------------------------------------------------------------------------
__global__ void scatter_kernel(const int* __restrict__ binOf,
                               int* __restrict__ cursors,
                               int* __restrict__ lists) {
  int b = blockIdx.x * blockDim.x + threadIdx.x;
  if (b >= BS_TOT) return;
  int bin = binOf[b];
  int pos = atomicAdd(&cursors[bin], 1);
  lists[pos] = b;
}

// ---------------------------------------------------------------------------
// Kernel 4: per (bin, 16-sample tile): D(16x128) = Zgather(16x128) @ L[bin]^T + m[bin]
// via chained V_WMMA_F32_16X16X4_F32.  256 threads = 8 wave32s; wave w owns
// output columns [16w, 16w+16).
// ---------------------------------------------------------------------------
#define LDS_PITCH 132   // 128 + 4 pad: lane stride 132 % 64 banks -> conflict-free frag reads

__global__ void __launch_bounds__(256)
gmm_wmma_kernel(const float* __restrict__ z,        // (BS, 128)
                const float* __restrict__ L,        // (C, K, 128, 128) row-major
                const float* __restrict__ mvec,     // (C, K, 128)
                const int*   __restrict__ counts,
                const int*   __restrict__ offsets,
                const int*   __restrict__ lists,
                float* __restrict__ out) {          // (BS, 128)
  __shared__ int   s_idx[16];
  __shared__ float s_A[16 * LDS_PITCH];

  const int bin       = blockIdx.x;
  const int cnt       = counts[bin];
  const int tileStart = blockIdx.y * 16;
  if (tileStart >= cnt) return;                     // uniform early exit (EXEC stays full)
  const int base = offsets[bin];

  const int tid = threadIdx.x;

  // Gather the 16 sample indices of this tile (pad invalid rows with sample 0;
  // their results are computed but never stored).
  if (tid < 16) {
    int sidx = 0;
    if (tileStart + tid < cnt) sidx = lists[base + tileStart + tid];
    s_idx[tid] = sidx;
  }
  __syncthreads();

  // Cooperative gather of A tile (16 x 128 f32) into LDS, float4 granularity.
  for (int e = tid; e < 16 * (N_Z / 4); e += 256) {
    int row = e >> 5;                               // 32 float4 per row
    int c4  = e & 31;
    float4 v = *((const float4*)(z + (size_t)s_idx[row] * N_Z) + c4);
    float* dst = &s_A[row * LDS_PITCH + c4 * 4];
    dst[0] = v.x; dst[1] = v.y; dst[2] = v.z; dst[3] = v.w;
  }
  __syncthreads();

  const int wave   = tid >> 5;                      // 0..7 -> output column tile
  const int lane   = tid & 31;
  const int laneLo = lane & 15;
  const int laneHi = lane >> 4;                     // 0 or 1
  const int n      = wave * 16 + laneLo;            // global output column 0..127

  // B[k][n] = L[bin][n][k]  (x = z @ L^T), so each lane walks row n of L[bin].
  const float* Lb = L + (size_t)bin * N_Z * N_Z + (size_t)n * N_Z + laneHi * 2;

  // C/D layout: VGPR r holds row (r + 8*laneHi), column n -> broadcast m[bin][n].
  const float mval = mvec[bin * N_Z + n];
  v8f acc = {mval, mval, mval, mval, mval, mval, mval, mval};

  // A layout: lane holds A[M = laneLo][K = kk + 2*laneHi + {0,1}].
  const float* aRow = &s_A[laneLo * LDS_PITCH + laneHi * 2];

#pragma unroll
  for (int kk = 0; kk < N_Z; kk += 4) {
    v2f a;  a.x  = aRow[kk];      a.y  = aRow[kk + 1];
    v2f bb; bb.x = Lb[kk];        bb.y = Lb[kk + 1];
    // 8 args: (neg_a, A, neg_b, B, c_mod, C, reuse_a, reuse_b)
    acc = __builtin_amdgcn_wmma_f32_16x16x4_f32(false, a, false, bb,
                                                (short)0, acc, false, false);
  }

  // Scatter results: VGPR r -> tile row (laneHi*8 + r), column n.
#pragma unroll
  for (int r = 0; r < 8; ++r) {
    int row = laneHi * 8 + r;
    if (tileStart + row < cnt) {
      out[(size_t)s_idx[row] * N_Z + n] = acc[r];
    }
  }
}

// ---------------------------------------------------------------------------
// Host launcher.  Inputs (setup_inputs order): z, y, gumbel, m, L, logits.
// Workspace layout (all ints, ~66 KB total):
//   [0,512)        counts[128]
//   [512,1024)     offsets[128]
//   [1024,1536)    cursors[128]
//   [1536,+32KB)   binOf[8192]
//   [+32KB,+64KB)  lists[8192]
// ---------------------------------------------------------------------------
extern "C" void kernel_launch(void* const* d_in, const int* in_sizes, int n_in,
                              void* d_out, int out_size, void* d_ws, size_t ws_size,
                              hipStream_t stream) {
  const float* z      = (const float*)d_in[0];
  const int*   y      = (const int*)d_in[1];   // JAX default x64-disabled -> int32
  const float* gumbel = (const float*)d_in[2];
  const float* m      = (const float*)d_in[3];
  const float* L      = (const float*)d_in[4];
  const float* logits = (const float*)d_in[5];
  float*       out    = (float*)d_out;

  char* ws = (char*)d_ws;
  int* counts  = (int*)(ws);
  int* offsets = (int*)(ws + 512);
  int* cursors = (int*)(ws + 1024);
  int* binOf   = (int*)(ws + 1536);
  int* lists   = (int*)(ws + 1536 + BS_TOT * sizeof(int));

  hipMemsetAsync(counts, 0, N_BINS * sizeof(int), stream);

  bin_count_kernel<<<BS_TOT / 256, 256, 0, stream>>>(gumbel, logits, y, counts, binOf);
  scan_kernel<<<1, 32, 0, stream>>>(counts, offsets, cursors);
  scatter_kernel<<<BS_TOT / 256, 256, 0, stream>>>(binOf, cursors, lists);

  // Worst case one bin holds all 8192 samples -> 512 tiles; empty tiles exit fast.
  dim3 grid(N_BINS, BS_TOT / 16, 1);
  gmm_wmma_kernel<<<grid, 256, 0, stream>>>(z, L, m, counts, offsets, lists, out);
}